// Hopfield_Kuramoto_multiplicative_75110388072808
// MI455X (gfx1250) — compile-verified
//
#include <hip/hip_runtime.h>
#include <hip/hip_bf16.h>

// ---------------------------------------------------------------------------
// Hopfield–Kuramoto multiplicative dynamics, MI455X (gfx1250, wave32).
//
// Roofline: W_hop is 8192x8192 f32 = 256 MB -> at 23.3 TB/s the single-pass
// symmetric matvec floor is ~11 us. Edge phase moves ~0.5 KB/edge * 262144
// edges ~ 270 MB but sk/fKacc (2 MB each) are L2-resident. Everything else
// is noise. Strategy: fuse W@g and W^T@g into ONE streaming pass over W,
// wave-per-edge gather/scatter with wave32 shuffle reductions, and do the
// sk @ (omega-omega^T)/2 GEMM with V_WMMA_F32_16X16X4_F32.
// ---------------------------------------------------------------------------

#define Nn 8192
#define Dd 64
#define Ee 262144

typedef float v2f __attribute__((ext_vector_type(2)));
typedef float v8f __attribute__((ext_vector_type(8)));

__device__ __forceinline__ float waveReduceSum(float v) {
  // wave32: 5-step butterfly, result valid in all lanes
  v += __shfl_xor(v, 16, 32);
  v += __shfl_xor(v, 8, 32);
  v += __shfl_xor(v, 4, 32);
  v += __shfl_xor(v, 2, 32);
  v += __shfl_xor(v, 1, 32);
  return v;
}

// ---------------------------------------------------------------------------
// K0: per-row normalize state_K -> sk ; g = tanh(state_H) ; f_H = -state_H ;
//     zero the f_K edge accumulator. One wave per row (lane owns a float2).
// ---------------------------------------------------------------------------
__global__ void hk_init(const float* __restrict__ state_H,
                        const float* __restrict__ state_K,
                        float* __restrict__ sk,
                        float* __restrict__ fKacc,
                        float* __restrict__ g,
                        float* __restrict__ fH /* d_out[0..N) */) {
  int lane = threadIdx.x & 31;
  int wave = threadIdx.x >> 5;
  int row = blockIdx.x * 8 + wave;
  if (row >= Nn) return;
  const float2* src = (const float2*)(state_K + (size_t)row * Dd);
  float2 v = src[lane];
  float ss = waveReduceSum(v.x * v.x + v.y * v.y);
  float rn = rsqrtf(ss);
  float2 o;
  o.x = v.x * rn;
  o.y = v.y * rn;
  ((float2*)(sk + (size_t)row * Dd))[lane] = o;
  float2 z;
  z.x = 0.f;
  z.y = 0.f;
  ((float2*)(fKacc + (size_t)row * Dd))[lane] = z;
  if (lane == 0) {
    float h = state_H[row];
    g[row] = tanhf(h);
    fH[row] = -h;
  }
}

// ---------------------------------------------------------------------------
// K1: fused symmetric matvec.  f_H += 0.5*(W@g + W^T@g), single pass over W.
// Block = 256 threads; tile = 256 rows x 1024 cols (thread owns 4 columns as
// float4).  Per row: 4 fma for the W^T@g register accumulator, a 4-wide dot
// + wave reduce + 1 LDS-atomic for the W@g row sum (cost amortized /4).
// Grid = (8 col strips, 32 row chunks) -> 256 blocks, 1 MB of W each.
// ---------------------------------------------------------------------------
__global__ void hk_wpass(const float* __restrict__ W,
                         const float* __restrict__ g,
                         float* __restrict__ fH) {
  __shared__ float rowsum[256];
  int tid = threadIdx.x;
  int lane = tid & 31;
  int c0 = blockIdx.x * 1024;
  int r0 = blockIdx.y * 256;
  rowsum[tid] = 0.f;
  __syncthreads();

  int c = c0 + 4 * tid;
  const float4 gc = *(const float4*)(g + c);
  float4 zacc;
  zacc.x = zacc.y = zacc.z = zacc.w = 0.f;

  for (int r = 0; r < 256; ++r) {
    int row = r0 + r;
    const float* Wp = W + (size_t)row * Nn + c;
    // prefetch the W stream a few rows ahead (gfx1250 global_prefetch_b8)
    if (r + 8 < 256) __builtin_prefetch(Wp + (size_t)8 * Nn, 0, 3);
    float4 w4 = *(const float4*)Wp;
    float gr = g[row];
    zacc.x = fmaf(w4.x, gr, zacc.x);
    zacc.y = fmaf(w4.y, gr, zacc.y);
    zacc.z = fmaf(w4.z, gr, zacc.z);
    zacc.w = fmaf(w4.w, gr, zacc.w);
    float rp = w4.x * gc.x + w4.y * gc.y + w4.z * gc.z + w4.w * gc.w;
    rp = waveReduceSum(rp);
    if (lane == 0) atomicAdd(&rowsum[r], rp);
  }
  __syncthreads();
  // W@g contribution for the 256 rows of this chunk (one atomic per row)
  unsafeAtomicAdd(&fH[r0 + tid], 0.5f * rowsum[tid]);
  // W^T@g contribution for this thread's 4 columns
  unsafeAtomicAdd(&fH[c + 0], 0.5f * zacc.x);
  unsafeAtomicAdd(&fH[c + 1], 0.5f * zacc.y);
  unsafeAtomicAdd(&fH[c + 2], 0.5f * zacc.z);
  unsafeAtomicAdd(&fH[c + 3], 0.5f * zacc.w);
}

// ---------------------------------------------------------------------------
// K2: ind_HK edges. One wave per edge; lane owns a float2 of the D=64 rows.
//   Gram = <sk_i, sk_j>;  w = (W[i,j]+W[j,i])/2
//   fH[i] += Gram*w*g_j/kH ; fH[j] += Gram*w*g_i/kH
//   fKacc[i] += -g_i*g_j*w/kK * sk_j ; symmetric for j
// ---------------------------------------------------------------------------
__global__ void hk_edges_HK(const int* __restrict__ indHK,
                            const float* __restrict__ W,
                            const float* __restrict__ sk,
                            const float* __restrict__ g,
                            const float* __restrict__ kHp,
                            const float* __restrict__ kKp,
                            float* __restrict__ fH,
                            float* __restrict__ fKacc) {
  int gid = blockIdx.x * blockDim.x + threadIdx.x;
  int e = gid >> 5;
  int lane = gid & 31;
  if (e >= Ee) return;
  int i = indHK[2 * e + 0];
  int j = indHK[2 * e + 1];
  float2 vi = ((const float2*)(sk + (size_t)i * Dd))[lane];
  float2 vj = ((const float2*)(sk + (size_t)j * Dd))[lane];
  float gram = waveReduceSum(vi.x * vj.x + vi.y * vj.y);
  float gi = g[i];
  float gj = g[j];
  float w = 0.5f * (W[(size_t)i * Nn + j] + W[(size_t)j * Nn + i]);
  float invKH = 1.0f / kHp[0];
  float invKK = 1.0f / kKp[0];
  if (lane == 0) unsafeAtomicAdd(&fH[i], gram * w * gj * invKH);
  if (lane == 1) unsafeAtomicAdd(&fH[j], gram * w * gi * invKH);
  float coef = -(gi * gj) * w * invKK;
  float* ai = fKacc + (size_t)i * Dd + 2 * lane;
  float* aj = fKacc + (size_t)j * Dd + 2 * lane;
  unsafeAtomicAdd(ai + 0, coef * vj.x);
  unsafeAtomicAdd(ai + 1, coef * vj.y);
  unsafeAtomicAdd(aj + 0, coef * vi.x);
  unsafeAtomicAdd(aj + 1, coef * vi.y);
}

// ---------------------------------------------------------------------------
// K3: ind_K edges.  s = <sk_a, sk_b>;  dE = c0*s + c1*s^2 + c2*s^3 + c3*s^4
//   fKacc[a] += dE*sk_b ; fKacc[b] += dE*sk_a
// ---------------------------------------------------------------------------
__global__ void hk_edges_K(const int* __restrict__ indK,
                           const float* __restrict__ sk,
                           const float* __restrict__ coeffs,
                           float* __restrict__ fKacc) {
  int gid = blockIdx.x * blockDim.x + threadIdx.x;
  int e = gid >> 5;
  int lane = gid & 31;
  if (e >= Ee) return;
  int a = indK[2 * e + 0];
  int b = indK[2 * e + 1];
  float2 va = ((const float2*)(sk + (size_t)a * Dd))[lane];
  float2 vb = ((const float2*)(sk + (size_t)b * Dd))[lane];
  float s = waveReduceSum(va.x * vb.x + va.y * vb.y);
  float c0 = coeffs[0], c1 = coeffs[1], c2 = coeffs[2], c3 = coeffs[3];
  float dE = s * fmaf(s, fmaf(s, fmaf(s, c3, c2), c1), c0);
  float* aa = fKacc + (size_t)a * Dd + 2 * lane;
  float* ab = fKacc + (size_t)b * Dd + 2 * lane;
  unsafeAtomicAdd(aa + 0, dE * vb.x);
  unsafeAtomicAdd(aa + 1, dE * vb.y);
  unsafeAtomicAdd(ab + 0, dE * va.x);
  unsafeAtomicAdd(ab + 1, dE * va.y);
}

// ---------------------------------------------------------------------------
// K4a: omega term via WMMA.  out_K = sk @ (omega - omega^T)/2  (8192x64 @ 64x64)
// One wave computes a 16-row slab, all 4 column tiles, with 16 K-steps of
// V_WMMA_F32_16X16X4_F32.  ISA VGPR layouts (CDNA5 ISA):
//   A 16x4 f32 : lanes 0-15 = rows, v0 holds K0 (lo half) / K2 (hi half),
//                v1 holds K1 / K3.
//   B 4x16 f32 : lanes 0-15 = cols, v0 = K0 / K2, v1 = K1 / K3.
//   C/D 16x16  : lane = col (mod 16), vgpr k = row k (lo half) / 8+k (hi).
// ---------------------------------------------------------------------------
__global__ void hk_omega_wmma(const float* __restrict__ sk,
                              const float* __restrict__ om,
                              float* __restrict__ fKout /* d_out + N */) {
  int lane = threadIdx.x & 31;
  int wave = threadIdx.x >> 5;
  int m0 = (blockIdx.x * 8 + wave) * 16;
  int half = lane >> 4;   // 0: K+0/K+1, 1: K+2/K+3
  int l15 = lane & 15;

  v8f acc0 = {}, acc1 = {}, acc2 = {}, acc3 = {};
  const float* skrow = sk + (size_t)(m0 + l15) * Dd;

  for (int kc = 0; kc < 16; ++kc) {
    int kb = 4 * kc + 2 * half;
    v2f A;
    A[0] = skrow[kb + 0];
    A[1] = skrow[kb + 1];
#pragma unroll
    for (int t = 0; t < 4; ++t) {
      int n = 16 * t + l15;
      v2f B;
      B[0] = 0.5f * (om[(kb + 0) * Dd + n] - om[n * Dd + (kb + 0)]);
      B[1] = 0.5f * (om[(kb + 1) * Dd + n] - om[n * Dd + (kb + 1)]);
      v8f* accp = (t == 0) ? &acc0 : (t == 1) ? &acc1 : (t == 2) ? &acc2 : &acc3;
      *accp = __builtin_amdgcn_wmma_f32_16x16x4_f32(
          /*neg_a=*/false, A, /*neg_b=*/false, B,
          /*c_mod=*/(short)0, *accp, /*reuse_a=*/false, /*reuse_b=*/false);
    }
  }
  // scatter D tiles: row = m0 + half*8 + k, col = 16*t + l15
#pragma unroll
  for (int t = 0; t < 4; ++t) {
    v8f acc = (t == 0) ? acc0 : (t == 1) ? acc1 : (t == 2) ? acc2 : acc3;
#pragma unroll
    for (int k = 0; k < 8; ++k) {
      int m = m0 + half * 8 + k;
      fKout[(size_t)m * Dd + 16 * t + l15] = acc[k];
    }
  }
}

// ---------------------------------------------------------------------------
// K4b: finalize.  f_K = -acc + sk * <sk,acc> + (omega term already in out).
// One wave per row.
// ---------------------------------------------------------------------------
__global__ void hk_finalize(const float* __restrict__ sk,
                            const float* __restrict__ fKacc,
                            float* __restrict__ fKout) {
  int lane = threadIdx.x & 31;
  int wave = threadIdx.x >> 5;
  int row = blockIdx.x * 8 + wave;
  if (row >= Nn) return;
  float2 s2 = ((const float2*)(sk + (size_t)row * Dd))[lane];
  float2 a2 = ((const float2*)(fKacc + (size_t)row * Dd))[lane];
  float proj = waveReduceSum(s2.x * a2.x + s2.y * a2.y);
  float2* op = (float2*)(fKout + (size_t)row * Dd) + lane;
  float2 o = *op;
  float2 r;
  r.x = -a2.x + s2.x * proj + o.x;
  r.y = -a2.y + s2.y * proj + o.y;
  *op = r;
}

// ---------------------------------------------------------------------------
extern "C" void kernel_launch(void* const* d_in, const int* in_sizes, int n_in,
                              void* d_out, int out_size, void* d_ws, size_t ws_size,
                              hipStream_t stream) {
  const float* state_H = (const float*)d_in[0];
  const float* state_K = (const float*)d_in[1];
  const float* W_hop = (const float*)d_in[2];
  const float* coeffs = (const float*)d_in[3];
  const float* omega = (const float*)d_in[4];
  const int* ind_K = (const int*)d_in[5];
  const int* ind_HK = (const int*)d_in[6];
  const float* kappa_K = (const float*)d_in[7];
  const float* kappa_H = (const float*)d_in[8];

  float* out = (float*)d_out;
  float* fH = out;            // N floats
  float* fK = out + Nn;       // N*D floats

  float* ws = (float*)d_ws;
  float* sk = ws;                       // N*D
  float* fKacc = ws + (size_t)Nn * Dd;  // N*D
  float* g = ws + 2 * (size_t)Nn * Dd;  // N

  // K0: normalize rows, tanh, init f_H, zero accumulator
  hk_init<<<Nn / 8, 256, 0, stream>>>(state_H, state_K, sk, fKacc, g, fH);

  // K1: single-pass fused symmetric matvec over W (256 MB streamed once)
  dim3 wgrid(Nn / 1024, Nn / 256);
  hk_wpass<<<wgrid, 256, 0, stream>>>(W_hop, g, fH);

  // K2/K3: wave-per-edge gather/scatter (L2-resident sk/fKacc)
  hk_edges_HK<<<Ee / 8, 256, 0, stream>>>(ind_HK, W_hop, sk, g, kappa_H,
                                          kappa_K, fH, fKacc);
  hk_edges_K<<<Ee / 8, 256, 0, stream>>>(ind_K, sk, coeffs, fKacc);

  // K4a: sk @ (omega-omega^T)/2 via V_WMMA_F32_16X16X4_F32
  hk_omega_wmma<<<Nn / (8 * 16), 256, 0, stream>>>(sk, omega, fK);

  // K4b: f_K = -acc + sk*<sk,acc> + omega term
  hk_finalize<<<Nn / 8, 256, 0, stream>>>(sk, fKacc, fK);
}